// Encoder_11802570130223
// MI455X (gfx1250) — compile-verified
//
#include <hip/hip_runtime.h>

typedef __attribute__((ext_vector_type(2))) float v2f;
typedef __attribute__((ext_vector_type(8))) float v8f;

#define NN 50000
#define NE 800000
#define DF 128
#define MTILES 3125  // 50000 / 16 exactly
#define GEMM_WAVES_PER_BLOCK 8
#define WT_PITCH 132  // transposed-W LDS pitch: even (b64 align) and ==4 mod 64 (bank-conflict-free b64 reads)

// ---------------- utility kernels ----------------

__global__ void k_fill_f32(float* __restrict__ p, long n, float v) {
    long i = (long)blockIdx.x * blockDim.x + threadIdx.x;
    if (i < n) p[i] = v;
}

__global__ void k_deg_accum(const int* __restrict__ col, float* __restrict__ deg, int e) {
    int i = blockIdx.x * blockDim.x + threadIdx.x;
    if (i < e) atomicAdd(&deg[col[i]], 1.0f);
}

__global__ void k_rsqrt_ip(float* __restrict__ d, int n) {
    int i = blockIdx.x * blockDim.x + threadIdx.x;
    if (i < n) d[i] = rsqrtf(d[i]);  // deg >= 1 always (self loop)
}

// ---------------- WMMA GEMM: H[M,128] = A[M,128] @ W[128,128] ----------------
// Block = 8 waves. W is staged once per block into LDS *transposed*
// (wt[n*132 + k]) via per-lane-addressed GLOBAL_LOAD_ASYNC_TO_LDS_B32 with
// purely incremental addressing (n is invariant per lane; k advances by 2 per
// step -> LDS addr += 8B, global addr += 1024B). B fragments for
// V_WMMA_F32_16X16X4_F32 are then single aligned bank-conflict-free
// ds_load_b64s straight into even VGPR pairs (compiler fuses them pairwise
// into ds_load_2addr_b64).
__global__ void __launch_bounds__(256)
k_gemm_wmma(const float* __restrict__ A, const float* __restrict__ W,
            float* __restrict__ H, int mtiles) {
    __shared__ float wt[DF * WT_PITCH];  // ~66 KB, transposed W

    // ---- async transpose-stage W -> LDS: 256 threads x 64 elements ----
    {
        unsigned tid = threadIdx.x;
        unsigned n0  = tid & 127u;          // W col (N): invariant per thread
        unsigned k0  = tid >> 7;            // W row (K) start: 0 or 1... (tid>>7 in {0,1})
        unsigned lds_addr = (unsigned)(size_t)&wt[0] + (n0 * WT_PITCH + k0) * 4u;
        unsigned long long gaddr = (unsigned long long)(const char*)W + (unsigned long long)tid * 4u;
#pragma unroll
        for (int it = 0; it < 64; ++it) {
            asm volatile("global_load_async_to_lds_b32 %0, %1, off"
                         :: "v"(lds_addr), "v"(gaddr) : "memory");
            lds_addr += 8u;       // k += 2
            gaddr    += 1024ull;  // g += 256 elements
        }
        asm volatile("s_wait_asynccnt 0x0" ::: "memory");
    }
    __syncthreads();

    int wave = (blockIdx.x * blockDim.x + threadIdx.x) >> 5;
    if (wave >= mtiles) return;  // after the barrier; whole wave exits together
    int lane = threadIdx.x & 31;
    int half = lane >> 4;
    int m15  = lane & 15;

    const float* arow = A + ((long)(wave * 16 + m15)) * DF;

    v8f acc[8];
#pragma unroll
    for (int nt = 0; nt < 8; ++nt) {
        v8f z = {0.f, 0.f, 0.f, 0.f, 0.f, 0.f, 0.f, 0.f};
        acc[nt] = z;
    }

#pragma unroll 4
    for (int kk = 0; kk < 32; ++kk) {
        int kbase = kk * 4 + half * 2;
        v2f a;
        a.x = arow[kbase + 0];
        a.y = arow[kbase + 1];
        // lane's transposed-W column base for this K pair
        const float* wcol = &wt[m15 * WT_PITCH + kbase];
#pragma unroll
        for (int nt = 0; nt < 8; ++nt) {
            v2f b = *(const v2f*)(wcol + nt * 16 * WT_PITCH);  // one ds_load_b64, 8B aligned
            acc[nt] = __builtin_amdgcn_wmma_f32_16x16x4_f32(
                false, a, false, b, (short)0, acc[nt], false, false);
        }
    }

    // C/D layout: VGPR v, lanes 0-15 -> M=v, N=lane; lanes 16-31 -> M=v+8
#pragma unroll
    for (int nt = 0; nt < 8; ++nt) {
#pragma unroll
        for (int v = 0; v < 8; ++v) {
            H[((long)(wave * 16 + v + half * 8)) * DF + nt * 16 + m15] = acc[nt][v];
        }
    }
}

// ---------------- edge scatter: AGG[c] += H[r] * dinv[r]*dinv[c] ----------------
// 128 threads per edge (feature-parallel, coalesced); atomics land in L2
// (per-layer working set ~83MB < 192MB L2).
__global__ void k_scatter(const int* __restrict__ row, const int* __restrict__ col,
                          const float* __restrict__ dinv, const float* __restrict__ H,
                          float* __restrict__ out, int e) {
    long idx = (long)blockIdx.x * blockDim.x + threadIdx.x;
    long ed  = idx >> 7;       // edge id
    int  t   = (int)(idx & 127);
    if (ed >= e) return;
    int r = row[ed], c = col[ed];
    float w = dinv[r] * dinv[c];
    atomicAdd(&out[(long)c * DF + t], H[(long)r * DF + t] * w);
}

// ---------------- epilogue: out = prelu(agg + H*dinv^2 + b, a) ----------------
__global__ void k_epilogue(const float* __restrict__ agg, const float* __restrict__ H,
                           const float* __restrict__ dinv, const float* __restrict__ b,
                           const float* __restrict__ a, float* __restrict__ out, int n) {
    long idx = (long)blockIdx.x * blockDim.x + threadIdx.x;
    long i   = idx >> 7;
    int  t   = (int)(idx & 127);
    if (i >= n) return;
    float di = dinv[i];
    float v  = agg[idx] + H[idx] * di * di + b[t];
    out[idx] = (v >= 0.f) ? v : a[t] * v;
}

// ---------------- launcher ----------------

extern "C" void kernel_launch(void* const* d_in, const int* in_sizes, int n_in,
                              void* d_out, int out_size, void* d_ws, size_t ws_size,
                              hipStream_t stream) {
    const float* x  = (const float*)d_in[0];
    const int*   ei = (const int*)  d_in[1];
    const float* W1 = (const float*)d_in[2];
    const float* b1 = (const float*)d_in[3];
    const float* a1 = (const float*)d_in[4];
    const float* W2 = (const float*)d_in[5];
    const float* b2 = (const float*)d_in[6];
    const float* a2 = (const float*)d_in[7];
    float* out = (float*)d_out;

    float* ws   = (float*)d_ws;
    float* dinv = ws;                       // 50000 floats (padded to 50176)
    float* H    = ws + 50176;               // 6,400,000 floats
    float* AGG  = H + (long)NN * DF;        // 6,400,000 floats

    const int* row = ei;           // edge_index[0]
    const int* col = ei + NE;      // edge_index[1]

    const long NELEM = (long)NN * DF;           // 6.4M
    const long ETHREADS = (long)NE * DF;        // 102.4M

    dim3 b256(256);
    // ---- degrees (shared by both layers) ----
    k_fill_f32<<<(NN + 255) / 256, b256, 0, stream>>>(dinv, NN, 1.0f);  // self-loop
    k_deg_accum<<<(NE + 255) / 256, b256, 0, stream>>>(col, dinv, NE);
    k_rsqrt_ip<<<(NN + 255) / 256, b256, 0, stream>>>(dinv, NN);

    int gemm_blocks = (MTILES + GEMM_WAVES_PER_BLOCK - 1) / GEMM_WAVES_PER_BLOCK;  // 391

    // ---- layer 1 ----
    k_fill_f32<<<(int)((NELEM + 255) / 256), b256, 0, stream>>>(AGG, NELEM, 0.0f);
    k_gemm_wmma<<<gemm_blocks, 256, 0, stream>>>(x, W1, H, MTILES);
    k_scatter<<<(int)((ETHREADS + 255) / 256), b256, 0, stream>>>(row, col, dinv, H, AGG, NE);
    k_epilogue<<<(int)((NELEM + 255) / 256), b256, 0, stream>>>(AGG, H, dinv, b1, a1, AGG, NN);

    // ---- layer 2 (AGG now holds layer-1 activations; d_out is AGG2) ----
    k_fill_f32<<<(int)((NELEM + 255) / 256), b256, 0, stream>>>(out, NELEM, 0.0f);
    k_gemm_wmma<<<gemm_blocks, 256, 0, stream>>>(AGG, W2, H, MTILES);
    k_scatter<<<(int)((ETHREADS + 255) / 256), b256, 0, stream>>>(row, col, dinv, H, out, NE);
    k_epilogue<<<(int)((NELEM + 255) / 256), b256, 0, stream>>>(out, H, dinv, b2, a2, out, NN);
}